// GATConvLayer_40123584479254
// MI455X (gfx1250) — compile-verified
//
#include <hip/hip_runtime.h>

typedef float v2f __attribute__((ext_vector_type(2)));
typedef float v8f __attribute__((ext_vector_type(8)));

#define INDIM 128
#define OUTD  64
#define HEADS 4
#define DHEAD 16

// ---------------------------------------------------------------------------
// Kernel 1: h[N,64] = x[N,128] @ W[64,128]^T using V_WMMA_F32_16X16X4_F32.
// One wave computes a 16x64 tile (4 accumulators of 16x16). K=128 -> 32 steps.
// W staged in LDS (padded rows: 16 B-frag lanes hit 16 distinct banks).
// A-row index is clamped instead of predicated: garbage rows only corrupt
// D rows that the guarded store discards, keeping the inner loop branchless.
// ---------------------------------------------------------------------------
__global__ __launch_bounds__(256) void gat_gemm_wmma(
    const float* __restrict__ x, const float* __restrict__ W,
    float* __restrict__ h, int N)
{
    __shared__ float Wlds[OUTD][INDIM + 2];   // 64 x 130 f32 = 33.3 KB

    for (int i = threadIdx.x; i < OUTD * INDIM; i += blockDim.x)
        Wlds[i >> 7][i & 127] = W[i];
    __syncthreads();

    const int wave = threadIdx.x >> 5;        // 8 waves -> 128 rows per block
    const int lane = threadIdx.x & 31;        // wave32
    const int half = lane >> 4;               // 0: K=0,1   1: K=2,3
    const int mrow = lane & 15;               // row (A) / col (B) within tile
    const int row0 = blockIdx.x * 128 + wave * 16;

    v8f acc0 = {}, acc1 = {}, acc2 = {}, acc3 = {};

    int arow = row0 + mrow;
    arow = arow < N ? arow : N - 1;           // clamp: branchless OOB handling
    const float* __restrict__ aptr = x + (size_t)arow * INDIM + half * 2;

    #pragma unroll 4
    for (int k = 0; k < INDIM; k += 4) {
        v2f a = *(const v2f*)(aptr + k);
        const int kk = k + half * 2;
        v2f b0 = *(const v2f*)&Wlds[mrow +  0][kk];
        v2f b1 = *(const v2f*)&Wlds[mrow + 16][kk];
        v2f b2 = *(const v2f*)&Wlds[mrow + 32][kk];
        v2f b3 = *(const v2f*)&Wlds[mrow + 48][kk];
        acc0 = __builtin_amdgcn_wmma_f32_16x16x4_f32(false, a, false, b0, (short)0, acc0, false, false);
        acc1 = __builtin_amdgcn_wmma_f32_16x16x4_f32(false, a, false, b1, (short)0, acc1, false, false);
        acc2 = __builtin_amdgcn_wmma_f32_16x16x4_f32(false, a, false, b2, (short)0, acc2, false, false);
        acc3 = __builtin_amdgcn_wmma_f32_16x16x4_f32(false, a, false, b3, (short)0, acc3, false, false);
    }

    // D layout: lane holds column N=mrow; VGPR r holds row M = r + 8*half.
    #pragma unroll
    for (int r = 0; r < 8; ++r) {
        const int row = row0 + r + 8 * half;
        if (row < N) {
            float* o = h + (size_t)row * OUTD + mrow;
            o[ 0] = acc0[r];
            o[16] = acc1[r];
            o[32] = acc2[r];
            o[48] = acc3[r];
        }
    }
}

// ---------------------------------------------------------------------------
// Kernel 2: per-node scores  score_{src,dst}[n,h] = <h[n,h,:], a_{src,dst}[h,:]>
// ---------------------------------------------------------------------------
__global__ void gat_scores(const float* __restrict__ h,
                           const float* __restrict__ a_src,
                           const float* __restrict__ a_dst,
                           float* __restrict__ ssrc,
                           float* __restrict__ sdst, int N)
{
    int t = blockIdx.x * blockDim.x + threadIdx.x;
    if (t >= N * HEADS) return;
    const int n  = t >> 2;
    const int hh = t & 3;
    const float* hp = h + (size_t)n * OUTD + hh * DHEAD;
    const float* as = a_src + hh * DHEAD;
    const float* ad = a_dst + hh * DHEAD;
    float s1 = 0.f, s2 = 0.f;
    #pragma unroll
    for (int d = 0; d < DHEAD; ++d) {
        float v = hp[d];
        s1 += v * as[d];
        s2 += v * ad[d];
    }
    ssrc[t] = s1;
    sdst[t] = s2;
}

// ---------------------------------------------------------------------------
// Kernel 3: edge scores + segment max.  e_max init = 0 implements
// max(segment_max, 0); only positive values can win, and positive IEEE floats
// order correctly as ints -> plain integer atomicMax.
// ---------------------------------------------------------------------------
__global__ void gat_edge_max(const int* __restrict__ src,
                             const int* __restrict__ dst,
                             const float* __restrict__ ssrc,
                             const float* __restrict__ sdst,
                             float* __restrict__ ebuf,
                             float* __restrict__ emax, int E)
{
    int e = blockIdx.x * blockDim.x + threadIdx.x;
    if (e >= E) return;
    const int s = src[e], d = dst[e];
    #pragma unroll
    for (int hh = 0; hh < HEADS; ++hh) {
        float v = ssrc[s * HEADS + hh] + sdst[d * HEADS + hh];
        v = v > 0.f ? v : 0.2f * v;                  // leaky_relu(0.2)
        ebuf[(size_t)e * HEADS + hh] = v;
        if (v > 0.f)
            atomicMax((int*)&emax[d * HEADS + hh], __float_as_int(v));
    }
}

// ---------------------------------------------------------------------------
// Kernel 4: e_exp = exp(e - e_max[dst]); segment sum via atomics.
// ---------------------------------------------------------------------------
__global__ void gat_edge_exp(const int* __restrict__ dst,
                             float* __restrict__ ebuf,
                             const float* __restrict__ emax,
                             float* __restrict__ esum, int E)
{
    int e = blockIdx.x * blockDim.x + threadIdx.x;
    if (e >= E) return;
    const int d = dst[e];
    #pragma unroll
    for (int hh = 0; hh < HEADS; ++hh) {
        float ex = __expf(ebuf[(size_t)e * HEADS + hh] - emax[d * HEADS + hh]);
        ebuf[(size_t)e * HEADS + hh] = ex;
        atomicAdd(&esum[d * HEADS + hh], ex);
    }
}

// ---------------------------------------------------------------------------
// Kernel 5: out[dst] += h[src] * alpha.  64 lanes per edge -> 256 B coalesced
// gather of h[src] and 256 B coalesced atomic burst to out[dst].
// ---------------------------------------------------------------------------
__global__ __launch_bounds__(256) void gat_aggregate(
    const int* __restrict__ src, const int* __restrict__ dst,
    const float* __restrict__ h, const float* __restrict__ ebuf,
    const float* __restrict__ esum, float* __restrict__ out, int E)
{
    const int e = blockIdx.x * (blockDim.x >> 6) + (threadIdx.x >> 6);
    if (e >= E) return;
    const int j  = threadIdx.x & 63;
    const int hh = j >> 4;
    const int s = src[e], d = dst[e];
    const float alpha = ebuf[(size_t)e * HEADS + hh] /
                        fmaxf(esum[d * HEADS + hh], 1e-9f);
    atomicAdd(&out[(size_t)d * OUTD + j], h[(size_t)s * OUTD + j] * alpha);
}

// ---------------------------------------------------------------------------
__global__ void zero_f32(float* __restrict__ p, long long n)
{
    long long i = (long long)blockIdx.x * blockDim.x + threadIdx.x;
    long long stride = (long long)gridDim.x * blockDim.x;
    for (; i < n; i += stride) p[i] = 0.f;
}

extern "C" void kernel_launch(void* const* d_in, const int* in_sizes, int n_in,
                              void* d_out, int out_size, void* d_ws, size_t ws_size,
                              hipStream_t stream)
{
    const float* x     = (const float*)d_in[0];
    const float* W     = (const float*)d_in[1];
    const float* a_src = (const float*)d_in[2];
    const float* a_dst = (const float*)d_in[3];
    const int*   ei    = (const int*)d_in[4];

    const int N = in_sizes[0] / INDIM;
    const int E = in_sizes[4] / 2;
    const int* src = ei;
    const int* dst = ei + E;
    float* out = (float*)d_out;

    // Workspace layout (floats): h | ssrc | sdst | emax | esum | ebuf  = ~56 MB
    float* h    = (float*)d_ws;
    float* ssrc = h    + (size_t)N * OUTD;
    float* sdst = ssrc + (size_t)N * HEADS;
    float* emax = sdst + (size_t)N * HEADS;
    float* esum = emax + (size_t)N * HEADS;
    float* ebuf = esum + (size_t)N * HEADS;

    zero_f32<<<1024, 256, 0, stream>>>(out,  (long long)N * OUTD);
    zero_f32<<<256,  256, 0, stream>>>(emax, (long long)N * HEADS);
    zero_f32<<<256,  256, 0, stream>>>(esum, (long long)N * HEADS);

    gat_gemm_wmma<<<(N + 127) / 128, 256, 0, stream>>>(x, W, h, N);
    gat_scores<<<(N * HEADS + 255) / 256, 256, 0, stream>>>(h, a_src, a_dst, ssrc, sdst, N);
    gat_edge_max<<<(E + 255) / 256, 256, 0, stream>>>(src, dst, ssrc, sdst, ebuf, emax, E);
    gat_edge_exp<<<(E + 255) / 256, 256, 0, stream>>>(dst, ebuf, emax, esum, E);
    gat_aggregate<<<(E + 3) / 4, 256, 0, stream>>>(src, dst, h, ebuf, esum, out, E);
}